// CensusLoss_24644522344666
// MI455X (gfx1250) — compile-verified
//
#include <hip/hip_runtime.h>
#include <hip/hip_bf16.h>
#include <stdint.h>

#define IMG_H 512
#define IMG_W 512
#define NIMG  24            // 8 * 3
#define TILE_W 32
#define TILE_H 64
#define HALO_W (TILE_W + 4) // 36
#define HALO_H (TILE_H + 4) // 68
#define HALO_N (HALO_W * HALO_H) // 2448
#define NBLK_X (IMG_W / TILE_W)  // 16
#define NBLK_Y (IMG_H / TILE_H)  // 8
#define NPART  (NBLK_X * NBLK_Y * NIMG) // 3072

typedef __attribute__((ext_vector_type(2))) float v2f;
typedef __attribute__((ext_vector_type(8))) float v8f;

__device__ __forceinline__ int reflect_idx(int x, int n) {
  // jnp.pad mode='reflect': -1 -> 1, -2 -> 2, n -> n-2, n+1 -> n-3
  x = (x < 0) ? -x : x;
  x = (x >= n) ? (2 * n - 2 - x) : x;
  return x;
}

__global__ __launch_bounds__(256) void census_partial(
    const float* __restrict__ pred, const float* __restrict__ gt,
    float* __restrict__ part) {
  __shared__ float sp[HALO_N];
  __shared__ float sg[HALO_N];
  __shared__ float wsum[8];

  const int tid  = threadIdx.y * 32 + threadIdx.x;
  const int tx   = threadIdx.x;          // column within tile: 0..31
  const int lr0  = threadIdx.y * 8;      // first of 8 output rows
  const int row0 = blockIdx.y * TILE_H;
  const int col0 = blockIdx.x * TILE_W;

  const float* pimg = pred + (size_t)blockIdx.z * (IMG_H * IMG_W);
  const float* gimg = gt   + (size_t)blockIdx.z * (IMG_H * IMG_W);

  // --- async gather of reflect-padded halo tile into LDS (CDNA5 path) ---
  for (int idx = tid; idx < HALO_N; idx += 256) {
    int r  = idx / HALO_W;
    int c  = idx - r * HALO_W;
    int gr = reflect_idx(row0 + r - 2, IMG_H);
    int gc = reflect_idx(col0 + c - 2, IMG_W);
    size_t off = (size_t)gr * IMG_W + gc;
    uint64_t pa = (uint64_t)(uintptr_t)(pimg + off);
    uint64_t ga = (uint64_t)(uintptr_t)(gimg + off);
    uint32_t lp = (uint32_t)(uintptr_t)&sp[idx];  // low 32 bits of flat LDS addr = LDS offset
    uint32_t lg = (uint32_t)(uintptr_t)&sg[idx];
    asm volatile("global_load_async_to_lds_b32 %0, %1, off"
                 :: "v"(lp), "v"(pa) : "memory");
    asm volatile("global_load_async_to_lds_b32 %0, %1, off"
                 :: "v"(lg), "v"(ga) : "memory");
  }
  asm volatile("s_wait_asynccnt 0x0" ::: "memory");
  __syncthreads();

  // --- registerize the 8 window centers for this thread's column ---
  float pc[8], gc8[8];
#pragma unroll
  for (int k = 0; k < 8; ++k) {
    pc[k]  = sp[(lr0 + 2 + k) * HALO_W + (tx + 2)];
    gc8[k] = sg[(lr0 + 2 + k) * HALO_W + (tx + 2)];
  }

  // --- census comparisons: ballot + popcount (wave-uniform accumulator) ---
  unsigned cnt = 0;
#pragma unroll
  for (int dj = 0; dj < 5; ++dj) {
#pragma unroll
    for (int dr = 0; dr < 12; ++dr) {
      float pv = sp[(lr0 + dr) * HALO_W + tx + dj];
      float gv = sg[(lr0 + dr) * HALO_W + tx + dj];
#pragma unroll
      for (int k = 0; k < 8; ++k) {
        int wr = dr - k;                 // window row for pixel k: needs 0..4
        if (wr < 0 || wr > 4) continue;
        if (dj == 2 && wr == 2) continue; // skip center
        unsigned mp = __builtin_amdgcn_ballot_w32(pv < pc[k]);
        unsigned mg = __builtin_amdgcn_ballot_w32(gv < gc8[k]);
        cnt += (unsigned)__builtin_popcount(mp ^ mg); // SALU, co-issues
      }
    }
  }

  const int lane = tid & 31;
  const int wave = tid >> 5;
  if (lane == 0) wsum[wave] = (float)cnt;  // cnt is wave-uniform via ballots
  __syncthreads();

  // --- combine the 8 wave sums with a real V_WMMA_F32_16X16X4_F32 ---
  // A = ones(16x4), B holds the 8 partials (rest 0) -> every D row = column sums.
  // Executed uniformly by ALL waves so EXEC is all-1s (WMMA requirement).
  v2f a; a.x = 1.0f; a.y = 1.0f;
  v2f b; b.x = (lane < 8) ? wsum[lane] : 0.0f; b.y = 0.0f;
  v8f cacc = {};
  v8f dres = __builtin_amdgcn_wmma_f32_16x16x4_f32(
      false, a, false, b, (short)0, cacc, false, false);
  float s = dres[0];                 // lanes 0..15: column sums of B
  s += __shfl_xor(s, 8, 32);
  s += __shfl_xor(s, 4, 32);
  s += __shfl_xor(s, 2, 32);
  s += __shfl_xor(s, 1, 32);         // lane 0: sum over lanes 0..15 = block total

  if (tid == 0) {
    int pidx = (blockIdx.z * NBLK_Y + blockIdx.y) * NBLK_X + blockIdx.x;
    part[pidx] = s;
  }
}

__global__ __launch_bounds__(256) void census_finish(
    const float* __restrict__ part, float* __restrict__ out) {
  __shared__ float red[256];
  const int tid = threadIdx.x;
  float s = 0.0f;
  for (int i = tid; i < NPART; i += 256) s += part[i];  // fixed order
  red[tid] = s;
  __syncthreads();
  for (int stride = 128; stride > 0; stride >>= 1) {
    if (tid < stride) red[tid] += red[tid + stride];
    __syncthreads();
  }
  if (tid == 0)
    out[0] = red[0] * (1.0f / (8.0f * 3.0f * 512.0f * 512.0f));
}

extern "C" void kernel_launch(void* const* d_in, const int* in_sizes, int n_in,
                              void* d_out, int out_size, void* d_ws, size_t ws_size,
                              hipStream_t stream) {
  const float* pred = (const float*)d_in[0];
  const float* gt   = (const float*)d_in[1];
  float* out  = (float*)d_out;
  float* part = (float*)d_ws;   // NPART floats = 12 KB of scratch

  dim3 grid(NBLK_X, NBLK_Y, NIMG);
  dim3 block(32, 8, 1);
  census_partial<<<grid, block, 0, stream>>>(pred, gt, part);
  census_finish<<<1, 256, 0, stream>>>(part, out);
}